// CRF_52982716563608
// MI455X (gfx1250) — compile-verified
//
#include <hip/hip_runtime.h>
#include <hip/hip_bf16.h>
#include <math.h>

// Problem constants from the reference (B, L, T) = (512, 512, 19)
#define Bsz 512
#define Lsz 512
#define Tn 19
#define KPAD 20          // K padded to multiple of 4 for 16x16x4 f32 WMMA
#define START_TAG 17
#define STOP_TAG 18

#define LOG2E 1.4426950408889634f   // log2(e)
#define LN2   0.6931471805599453f   // ln(2)

typedef __attribute__((ext_vector_type(2))) float v2f;
typedef __attribute__((ext_vector_type(8))) float v8f;

// ---------------------------------------------------------------------------
// Forward (log-partition) kernel: one wave handles 16 batch rows.
// p-recurrence done in exp-domain so the 16x19 @ 19x19 inner product maps to
// v_wmma_f32_16x16x4_f32 (5 K-chunks x 2 N-tiles = 10 WMMAs per step).
//
// The scan state q = p * log2(e) is kept in BASE-2 log domain so the per-step
// transcendentals are bare v_exp_f32 / v_log_f32 (the hardware's native
// base-2 ops) with zero range-correction scaffolding on the critical chain:
//   exp2(q - Mq) == exp(p - Mp)        (identical exp-domain value)
//   q_new = f*log2e + Mq + log2(acc)   (one FMA folds the feats term)
// Conversion back to natural log happens once, at the final reduction.
// ---------------------------------------------------------------------------
__global__ __launch_bounds__(32) void crf_forward_wmma(
    const float* __restrict__ feats,   // [B, L, T]
    const float* __restrict__ trans,   // [T, T]
    float* __restrict__ partials)      // [B/16] per-wave partial log_z sums
{
  // 16 real rows + 1 dummy row (row 16) so invalid tile-1 lanes can store
  // unconditionally (address cndmask instead of exec-mask branches).
  __shared__ float lds[17 * KPAD];

  const int lane = threadIdx.x;        // 0..31
  const int nl   = lane & 15;          // position within 16-lane half
  const int half = lane >> 4;          // 0 or 1
  const int b0   = blockIdx.x * 16;    // first batch row of this wave
  const int j1c  = (16 + nl < Tn) ? (16 + nl) : 0;  // clamped tile-1 tag index
  const bool j1v = (16 + nl < Tn);                  // tile-1 tag validity

  // zero the K-pad column once (never overwritten by the step loop)
  if (lane < 16) lds[lane * KPAD + 19] = 0.0f;
  __syncthreads();

  // --- Precompute B operands: expT = exp(transitions), padded with zeros ---
  // One-time cost: keep full-accuracy expf here (constants reused 512x).
  // NOTE: natural exp is correct here: exp2(q-Mq)*exp(trans) is exactly the
  // exp-domain product regardless of which log base the scan state uses.
  // B (4x16 KxN) per-chunk layout (mirrors documented A layout):
  //   .x -> K = 4c + 2*half ; .y -> K = 4c + 2*half + 1 ; N = tile*16 + nl
  v2f Bm[2][5];
#pragma unroll
  for (int t = 0; t < 2; ++t) {
#pragma unroll
    for (int c = 0; c < 5; ++c) {
      const int j  = t * 16 + nl;
      const int r0 = 4 * c + 2 * half;
      const int r1 = r0 + 1;
      const int jc = (j < Tn) ? j : 0;
      float bx = expf(trans[((r0 < Tn) ? r0 : 0) * Tn + jc]);
      float by = expf(trans[((r1 < Tn) ? r1 : 0) * Tn + jc]);
      if (!(r0 < Tn && j < Tn)) bx = 0.0f;
      if (!(r1 < Tn && j < Tn)) by = 0.0f;
      v2f b; b.x = bx; b.y = by;
      Bm[t][c] = b;
    }
  }

  // --- q state (base-2 log domain) kept in D/C layout:
  //     pt0[r] = q[batch r+8*half][tag nl],
  //     pt1[r] = q[batch r+8*half][tag 16+nl] (invalid tags -> -inf) ---
  float pt0[8], pt1[8];
#pragma unroll
  for (int r = 0; r < 8; ++r) {
    const int b = b0 + r + 8 * half;
    const float* fb = feats + (size_t)b * (Lsz * Tn);  // l = 0
    pt0[r] = (fb[nl] + trans[START_TAG * Tn + nl]) * LOG2E;
    const float v1 = (fb[j1c] + trans[START_TAG * Tn + j1c]) * LOG2E;
    pt1[r] = j1v ? v1 : -INFINITY;                     // value cndmask
  }

  const int t1dummy = 16 * KPAD;       // dummy LDS row for invalid t1 lanes

  float Mr[8];
  v2f A[5];

  for (int l = 1; l < Lsz; ++l) {
    // 1) per-batch max over tags (butterfly within each 16-lane half)
#pragma unroll
    for (int r = 0; r < 8; ++r) {
      float v = fmaxf(pt0[r], pt1[r]);       // invalid tile1 lanes are -inf
      v = fmaxf(v, __shfl_xor(v, 1, 32));
      v = fmaxf(v, __shfl_xor(v, 2, 32));
      v = fmaxf(v, __shfl_xor(v, 4, 32));
      v = fmaxf(v, __shfl_xor(v, 8, 32));
      Mr[r] = v;                              // replicated across the half
    }

    // 2) expP = exp2(q - Mq), transpose D-layout -> A-layout via LDS
    __syncthreads();
#pragma unroll
    for (int r = 0; r < 8; ++r) {
      const int m = r + 8 * half;             // batch row within tile
      lds[m * KPAD + nl] = __builtin_amdgcn_exp2f(pt0[r] - Mr[r]);
      // tile-1: exp2(-inf)=0 for invalid lanes; steer them to dummy row 16
      const int soff = (nl < 3) ? (m * KPAD + 16 + nl) : (t1dummy + nl);
      lds[soff] = __builtin_amdgcn_exp2f(pt1[r] - Mr[r]);
    }
    __syncthreads();
#pragma unroll
    for (int c = 0; c < 5; ++c) {             // A row M = nl; K split per half
      v2f a;
      a.x = lds[nl * KPAD + 4 * c + 2 * half];
      a.y = lds[nl * KPAD + 4 * c + 2 * half + 1];
      A[c] = a;
    }

    // 3) D = expP(16x20) x expT(20x19) via chained WMMA accumulation
    v8f acc0 = {}; v8f acc1 = {};
#pragma unroll
    for (int c = 0; c < 5; ++c)
      acc0 = __builtin_amdgcn_wmma_f32_16x16x4_f32(
          false, A[c], false, Bm[0][c], (short)0, acc0, false, false);
#pragma unroll
    for (int c = 0; c < 5; ++c)
      acc1 = __builtin_amdgcn_wmma_f32_16x16x4_f32(
          false, A[c], false, Bm[1][c], (short)0, acc1, false, false);

    // 4) q_new = feats[l]*log2e + Mq + log2(D)
    //    tile-1 feats load is address-clamped and unconditional: for invalid
    //    columns log2(acc1) = log2(0) = -inf dominates, so the value is moot.
#pragma unroll
    for (int r = 0; r < 8; ++r) {
      const int b = b0 + r + 8 * half;
      const float* fb = feats + ((size_t)b * Lsz + l) * Tn;
      const float f0 = fb[nl];
      const float f1 = fb[j1c];
      pt0[r] = fmaf(f0, LOG2E, Mr[r] + __builtin_amdgcn_logf(acc0[r]));
      pt1[r] = fmaf(f1, LOG2E, Mr[r] + __builtin_amdgcn_logf(acc1[r]));
    }
  }

  // --- Epilogue: final = Mq + log2(expP x expT); take column STOP (=18) ---
#pragma unroll
  for (int r = 0; r < 8; ++r) {
    float v = fmaxf(pt0[r], pt1[r]);
    v = fmaxf(v, __shfl_xor(v, 1, 32));
    v = fmaxf(v, __shfl_xor(v, 2, 32));
    v = fmaxf(v, __shfl_xor(v, 4, 32));
    v = fmaxf(v, __shfl_xor(v, 8, 32));
    Mr[r] = v;
  }
  __syncthreads();
#pragma unroll
  for (int r = 0; r < 8; ++r) {
    const int m = r + 8 * half;
    lds[m * KPAD + nl] = __builtin_amdgcn_exp2f(pt0[r] - Mr[r]);
    const int soff = (nl < 3) ? (m * KPAD + 16 + nl) : (t1dummy + nl);
    lds[soff] = __builtin_amdgcn_exp2f(pt1[r] - Mr[r]);
  }
  __syncthreads();
#pragma unroll
  for (int c = 0; c < 5; ++c) {
    v2f a;
    a.x = lds[nl * KPAD + 4 * c + 2 * half];
    a.y = lds[nl * KPAD + 4 * c + 2 * half + 1];
    A[c] = a;
  }
  v8f acc1 = {};
#pragma unroll
  for (int c = 0; c < 5; ++c)
    acc1 = __builtin_amdgcn_wmma_f32_16x16x4_f32(
        false, A[c], false, Bm[1][c], (short)0, acc1, false, false);

  // STOP tag = 18 lives in tile1 at nl == 2 (lanes 2 and 18)
  float s = 0.0f;
#pragma unroll
  for (int r = 0; r < 8; ++r) s += Mr[r] + __builtin_amdgcn_logf(acc1[r]);
  s += __shfl_xor(s, 16, 32);                 // lane2 (+half0) + lane18 (+half1)
  s *= LN2;                                   // base-2 -> natural log, once
  if (lane == 2) partials[blockIdx.x] = s;
}

// ---------------------------------------------------------------------------
// Viterbi kernel: one wave per batch; lane = current tag j (lanes 19..31 pad).
// Tropical semiring -> scalar VALU; p[i] broadcast via v_readlane shuffles.
// Backpointers stored as u8; lane 0 backtraces in-wave after s_wait_storecnt.
// ---------------------------------------------------------------------------
__global__ __launch_bounds__(256) void crf_viterbi(
    const float* __restrict__ feats,   // [B, L, T]
    const float* __restrict__ trans,   // [T, T]
    unsigned char* __restrict__ bp,    // [B, L, T] workspace backpointers
    float* __restrict__ out_dec)       // d_out + 1 : decode as floats [B, L]
{
  const int wave = threadIdx.x >> 5;
  const int lane = threadIdx.x & 31;
  const int b = blockIdx.x * 8 + wave;
  const bool jvalid = (lane < Tn);
  const int j = jvalid ? lane : (Tn - 1);    // clamp pad lanes (unused results)

  float tcol[Tn];                            // trans[:, j] column in registers
#pragma unroll
  for (int i = 0; i < Tn; ++i) tcol[i] = trans[i * Tn + j];

  const float* fb = feats + (size_t)b * (Lsz * Tn);
  unsigned char* bpb = bp + (size_t)b * (Lsz * Tn);

  float pv = fb[j] + trans[START_TAG * Tn + j];   // p0

  for (int l = 1; l < Lsz; ++l) {
    float best = -INFINITY; int bi = 0;
#pragma unroll
    for (int i = 0; i < Tn; ++i) {
      const float cur = __shfl(pv, i, 32) + tcol[i];
      if (cur > best) { best = cur; bi = i; } // first-max tie rule (== argmax)
    }
    if (jvalid) bpb[(size_t)l * Tn + j] = (unsigned char)bi;
    pv = best + fb[(size_t)l * Tn + j];
  }

  // pointer = argmax_i( p_last[i] + trans[i][STOP] )
  float best = -INFINITY; int ptr = 0;
#pragma unroll
  for (int i = 0; i < Tn; ++i) {
    const float cur = __shfl(pv, i, 32) + trans[i * Tn + STOP_TAG];
    if (cur > best) { best = cur; ptr = i; }
  }

  // make our own BP stores visible before the pointer chase (same wave)
  asm volatile("s_wait_storecnt 0x0" ::: "memory");

  if (lane == 0) {
    float* db = out_dec + (size_t)b * Lsz;
    db[Lsz - 1] = (float)ptr;
    int cur = ptr;
    for (int l = Lsz - 1; l >= 1; --l) {
      cur = bpb[(size_t)l * Tn + cur];
      db[l - 1] = (float)cur;
    }
  }
}

// Deterministic fixed-order reduction of the 32 per-wave partial sums.
__global__ void crf_reduce(const float* __restrict__ partials,
                           float* __restrict__ out) {
  if (threadIdx.x == 0) {
    float s = 0.0f;
    for (int i = 0; i < Bsz / 16; ++i) s += partials[i];
    out[0] = s;
  }
}

// ---------------------------------------------------------------------------
// d_in[0] = feats (B,L,T) f32 ; d_in[1] = mask (B,L) bool (all true, lengths=L)
// d_in[2] = transitions (T,T) f32
// d_out: [ log_z (1 float) | decode (B*L floats, row-major) ]
// d_ws : [ 32 partial sums (512 B region) | BP u8 [B,L,T] (~4.98 MB) ]
// ---------------------------------------------------------------------------
extern "C" void kernel_launch(void* const* d_in, const int* in_sizes, int n_in,
                              void* d_out, int out_size, void* d_ws, size_t ws_size,
                              hipStream_t stream) {
  const float* feats = (const float*)d_in[0];
  const float* trans = (const float*)d_in[2];
  float* out = (float*)d_out;

  float* partials = (float*)d_ws;
  unsigned char* bp = (unsigned char*)d_ws + 512;

  crf_forward_wmma<<<Bsz / 16, 32, 0, stream>>>(feats, trans, partials);
  crf_viterbi<<<Bsz / 8, 256, 0, stream>>>(feats, trans, bp, out + 1);
  crf_reduce<<<1, 32, 0, stream>>>(partials, out);
}